// customContrastiveLoss_75342316306685
// MI455X (gfx1250) — compile-verified
//
#include <hip/hip_runtime.h>

// CDNA5 / gfx1250, wave32. Bandwidth-bound contrastive-loss reduction.
// Core op: per-batch (16x512)·(512) dot products via V_WMMA_F32_16X16X4_F32,
// one wave per (matrix, batch) pair, accumulating over D in chunks of 4.

typedef __attribute__((ext_vector_type(2))) float v2f;
typedef __attribute__((ext_vector_type(8))) float v8f;

#define B_SZ 4096
#define K_SZ 16
#define D_SZ 512

// ---------------------------------------------------------------------------
// Pass 1: numerator terms. One wave per batch: exp(<s_b, t_b>) -> ws[2B + b]
// ---------------------------------------------------------------------------
__global__ __launch_bounds__(256) void numerator_kernel(
    const float* __restrict__ s, const float* __restrict__ t,
    float* __restrict__ ws) {
  const int lane = threadIdx.x & 31;
  const int wv   = threadIdx.x >> 5;
  const int b    = blockIdx.x * 8 + wv;        // 0..4095

  const float4* sp = (const float4*)(s + (size_t)b * D_SZ);
  const float4* tp = (const float4*)(t + (size_t)b * D_SZ);

  float acc = 0.f;
#pragma unroll
  for (int it = 0; it < D_SZ / (32 * 4); ++it) {   // 4 iterations
    float4 a = sp[it * 32 + lane];
    float4 c = tp[it * 32 + lane];
    acc += a.x * c.x + a.y * c.y + a.z * c.z + a.w * c.w;
  }
#pragma unroll
  for (int off = 16; off > 0; off >>= 1) acc += __shfl_xor(acc, off);

  if (lane == 0) ws[2 * B_SZ + b] = __expf(acc);
}

// ---------------------------------------------------------------------------
// Pass 2: denominators via WMMA. Wave w handles (sel = w>>12, b = w&4095):
//   sel==0: den_NS[b] = sum_k exp(<NS[b,k], t[b]>)  -> ws[b]       (log)
//   sel==1: den_NT[b] = sum_k exp(<NT[b,k], s[b]>)  -> ws[B + b]   (log)
// A = 16x4 tile of the neighbor matrix, B = 4 t-values broadcast to 16 cols.
// f32 WMMA A-layout: lane L holds A[L&15, 2*(L>>4) + {0,1}] in 2 VGPRs ->
// a single b64 load of two consecutive floats per lane per step.
// ---------------------------------------------------------------------------
__global__ __launch_bounds__(256) void denominator_kernel(
    const float* __restrict__ s, const float* __restrict__ t,
    const float* __restrict__ NS, const float* __restrict__ NT,
    float* __restrict__ ws) {
  const int lane = threadIdx.x & 31;
  const int wv   = threadIdx.x >> 5;
  const int w    = blockIdx.x * 8 + wv;        // 0..8191
  const int sel  = w >> 12;                    // 0 -> NS·t, 1 -> NT·s
  const int b    = w & (B_SZ - 1);

  const float* M = (sel ? NT : NS) + (size_t)b * (K_SZ * D_SZ);
  const float* V = (sel ? s  : t)  + (size_t)b * D_SZ;

  const int m   = lane & 15;           // A-matrix row owned by this lane
  const int kkb = (lane >> 4) << 1;    // K sub-offset: 0 for lanes 0-15, 2 for 16-31

  const float* arow = M + m * D_SZ + kkb;
  const float* bvec = V + kkb;

  v8f acc = {};
#pragma unroll 4
  for (int d0 = 0; d0 < D_SZ; d0 += 4) {
    v2f av = *(const v2f*)(arow + d0);       // global_load_b64 (A tile)
    v2f bv = *(const v2f*)(bvec + d0);       // global_load_b64 (broadcast B), L0 hit
    // 8 args: (neg_a, A, neg_b, B, c_mod, C, reuse_a, reuse_b)
    acc = __builtin_amdgcn_wmma_f32_16x16x4_f32(
        false, av, false, bv, (short)0, acc, false, false);
  }

  // D[m,n] identical across n. Lanes 0-15 hold M=0..7, lanes 16-31 hold M=8..15.
  float part = 0.f;
#pragma unroll
  for (int r = 0; r < 8; ++r) part += __expf(acc[r]);
  part += __shfl_xor(part, 16);              // full sum over 16 rows

  if (lane == 0) ws[sel * B_SZ + b] = __logf(part);
}

// ---------------------------------------------------------------------------
// Pass 3: deterministic finalize. Single block reduces the three 4096-vectors
// and emits (-log_exp_s_t, -log_exp_t_s).
// ---------------------------------------------------------------------------
__global__ __launch_bounds__(256) void finalize_kernel(
    const float* __restrict__ ws, float* __restrict__ out) {
  __shared__ float red[3][8];
  const int tid  = threadIdx.x;
  const int lane = tid & 31;
  const int wv   = tid >> 5;

  float a0 = 0.f, a1 = 0.f, a2 = 0.f;
  for (int i = tid; i < B_SZ; i += 256) {
    a0 += ws[i];                 // sum_b log(den_NS[b])
    a1 += ws[B_SZ + i];          // sum_b log(den_NT[b])
    a2 += ws[2 * B_SZ + i];      // numerator = sum_b exp(<s_b,t_b>)
  }
#pragma unroll
  for (int off = 16; off > 0; off >>= 1) {
    a0 += __shfl_xor(a0, off);
    a1 += __shfl_xor(a1, off);
    a2 += __shfl_xor(a2, off);
  }
  if (lane == 0) { red[0][wv] = a0; red[1][wv] = a1; red[2][wv] = a2; }
  __syncthreads();

  if (tid == 0) {
    float S_NS = 0.f, S_NT = 0.f, num = 0.f;
#pragma unroll
    for (int i = 0; i < 8; ++i) {
      S_NS += red[0][i]; S_NT += red[1][i]; num += red[2][i];
    }
    const float logn = __logf(num);
    const float log_exp_t_s = ((float)B_SZ * logn - S_NS) * (1.0f / (float)K_SZ);
    const float log_exp_s_t = ((float)B_SZ * logn - S_NT) * (1.0f / (float)K_SZ);
    out[0] = -log_exp_s_t;
    out[1] = -log_exp_t_s;
  }
}

extern "C" void kernel_launch(void* const* d_in, const int* in_sizes, int n_in,
                              void* d_out, int out_size, void* d_ws, size_t ws_size,
                              hipStream_t stream) {
  const float* s  = (const float*)d_in[0];
  const float* t  = (const float*)d_in[1];
  const float* NS = (const float*)d_in[2];
  const float* NT = (const float*)d_in[3];
  float* ws  = (float*)d_ws;   // 3 * 4096 floats = 48 KB, fully rewritten each call
  float* out = (float*)d_out;  // 2 floats

  numerator_kernel  <<<B_SZ / 8,     256, 0, stream>>>(s, t, ws);
  denominator_kernel<<<2 * B_SZ / 8, 256, 0, stream>>>(s, t, NS, NT, ws);
  finalize_kernel   <<<1,            256, 0, stream>>>(ws, out);
}